// BrainAgeGATv2_36893769072795
// MI455X (gfx1250) — compile-verified
//
#include <hip/hip_runtime.h>

// ---------------------------------------------------------------------------
// BrainAge GATv2 on MI455X (gfx1250, wave32, WMMA).
// GEMMs use V_WMMA_F32_16X16X4_F32 (full fp32 precision). W is staged in LDS
// k-pair-interleaved so each B fragment is a single aligned ds_load_b64.
// Edge softmax / scatter passes are HBM-bound: coalesced float4 + f32 atomics.
// ---------------------------------------------------------------------------

#define N_NODES 25600
#define NPG     200
#define G_GR    128
#define E_EDGES 614400
#define ET_EDGES (E_EDGES + N_NODES)   // with self loops
#define GF_DIM  16
#define HC      128                    // H*C
#define NEG_SLOPE 0.2f
#define BN_EPS  1e-5f

typedef float v2f __attribute__((ext_vector_type(2)));
typedef float v8f __attribute__((ext_vector_type(8)));

// ---------------------------------------------------------------- utilities
__global__ void fill_kernel(float* __restrict__ p, float v, int n) {
  int i = blockIdx.x * blockDim.x + threadIdx.x;
  if (i < n) p[i] = v;
}

__device__ __forceinline__ void atomicMaxF(float* addr, float v) {
  // Ordered-int encoding trick; valid with -inf initialization.
  if (v >= 0.0f) atomicMax((int*)addr, __float_as_int(v));
  else           atomicMin((unsigned int*)addr, __float_as_uint(v));
}

// ---------------------------------------------------------------- embedding
// h0 = relu(x @ We + be), x:[N,3], We:[3,64]
__global__ void embed_kernel(const float* __restrict__ x,
                             const float* __restrict__ We,
                             const float* __restrict__ be,
                             float* __restrict__ h) {
  int i = blockIdx.x * blockDim.x + threadIdx.x;
  if (i >= N_NODES * 64) return;
  int n = i >> 6, j = i & 63;
  float v = be[j] + x[n * 3 + 0] * We[j]
                  + x[n * 3 + 1] * We[64 + j]
                  + x[n * 3 + 2] * We[128 + j];
  h[i] = v > 0.0f ? v : 0.0f;
}

// ---------------------------------------------------------------- WMMA GEMM
// out[N,128] = h[N,DIN] @ W[DIN,128] + bias.
// Block = 256 thr (8 waves) -> 128x128 tile. Wave w: rows [w*16, w*16+16).
// A frag (16x4 f32): lane L holds A[L&15][2*(L>>4) + {0,1}]  (float2 load).
// B frag (4x16 f32): lane L holds B[2*(L>>4) + {0,1}][L&15].
//   LDS holds W k-pair-interleaved: sW[p*256 + c*2 + {0,1}] = W[2p][c],W[2p+1][c]
//   -> fragment = one aligned ds_load_b64, no repacking moves.
// C/D (16x16 f32):   VGPR i, lanes 0-15 -> M=i, lanes 16-31 -> M=i+8.
template <int DIN>
__global__ __launch_bounds__(256) void wmma_gemm_kernel(
    const float* __restrict__ h, const float* __restrict__ W,
    const float* __restrict__ bias, float* __restrict__ out) {
  __shared__ float sW[DIN * 128];

  const int tid = threadIdx.x;
  // stage W into LDS, k-pair interleaved (8 elements per thread-step)
  for (int idx = tid; idx < DIN * 16; idx += 256) {
    const int p  = idx >> 5;          // k-pair index (0..DIN/2-1)
    const int cg = (idx & 31) * 4;    // column group base (0..124)
    const float4 w0 = *(const float4*)&W[(size_t)(2 * p) * 128 + cg];
    const float4 w1 = *(const float4*)&W[(size_t)(2 * p + 1) * 128 + cg];
    float4 lo, hi;
    lo.x = w0.x; lo.y = w1.x; lo.z = w0.y; lo.w = w1.y;
    hi.x = w0.z; hi.y = w1.z; hi.z = w0.w; hi.w = w1.w;
    *(float4*)&sW[p * 256 + cg * 2]     = lo;
    *(float4*)&sW[p * 256 + cg * 2 + 4] = hi;
  }
  __syncthreads();

  const int wave = tid >> 5;
  const int lane = tid & 31;
  const int m    = lane & 15;   // row of A-tile / col of B-tile
  const int g    = lane >> 4;   // K-half selector
  const int rowBase = blockIdx.x * 128 + wave * 16;

  v8f acc[8] = {};

  const float* arow = h + (size_t)(rowBase + m) * DIN;

  for (int k = 0; k < DIN; k += 4) {
    const float2 av = *(const float2*)&arow[k + 2 * g];
    v2f a; a.x = av.x; a.y = av.y;
    // rows (k+2g, k+2g+1) => pair index k/2+g; col t*16+m => offset (t*16+m)*2
    const float* sb = &sW[((k >> 1) + g) * 256 + m * 2];
#pragma unroll
    for (int t = 0; t < 8; ++t) {
      const v2f b = *(const v2f*)&sb[t * 32];
      acc[t] = __builtin_amdgcn_wmma_f32_16x16x4_f32(
          false, a, false, b, (short)0, acc[t], false, false);
    }
  }

#pragma unroll
  for (int t = 0; t < 8; ++t) {
    const float bcol = bias[t * 16 + m];
#pragma unroll
    for (int i = 0; i < 8; ++i) {
      const int row = rowBase + i + 8 * g;
      out[(size_t)row * HC + t * 16 + m] = acc[t][i] + bcol;
    }
  }
}

// ---------------------------------------------------------------- edge pass 1
// logit[e,h] = sum_c leakyrelu(xl[src]+xr[dst]) * att[h,c]; segment max by dst
__global__ void edge_logit_kernel(const float* __restrict__ xl,
                                  const float* __restrict__ xr,
                                  const float* __restrict__ att,
                                  const int* __restrict__ ei,
                                  float* __restrict__ logit,
                                  float* __restrict__ mx) {
  int tid = blockIdx.x * blockDim.x + threadIdx.x;
  if (tid >= ET_EDGES * 8) return;
  const int e = tid >> 3, hh = tid & 7;
  const int s = (e < E_EDGES) ? ei[e]            : (e - E_EDGES);
  const int d = (e < E_EDGES) ? ei[E_EDGES + e]  : (e - E_EDGES);
  const float4* pl = (const float4*)(xl + (size_t)s * HC + hh * 16);
  const float4* pr = (const float4*)(xr + (size_t)d * HC + hh * 16);
  const float4* pa = (const float4*)(att + hh * 16);
  float acc = 0.0f;
#pragma unroll
  for (int q = 0; q < 4; ++q) {
    const float4 l = pl[q], r = pr[q], a = pa[q];
    float v;
    v = l.x + r.x; v = v > 0.0f ? v : NEG_SLOPE * v; acc += v * a.x;
    v = l.y + r.y; v = v > 0.0f ? v : NEG_SLOPE * v; acc += v * a.y;
    v = l.z + r.z; v = v > 0.0f ? v : NEG_SLOPE * v; acc += v * a.z;
    v = l.w + r.w; v = v > 0.0f ? v : NEG_SLOPE * v; acc += v * a.w;
  }
  logit[tid] = acc;
  atomicMaxF(&mx[d * 8 + hh], acc);
}

// ---------------------------------------------------------------- edge pass 2
// a = exp(logit - mx[dst]); den[dst] += a  (a overwrites logit buffer)
__global__ void edge_exp_kernel(float* __restrict__ lg,
                                const float* __restrict__ mx,
                                float* __restrict__ den,
                                const int* __restrict__ ei) {
  int tid = blockIdx.x * blockDim.x + threadIdx.x;
  if (tid >= ET_EDGES * 8) return;
  const int e = tid >> 3, hh = tid & 7;
  const int d = (e < E_EDGES) ? ei[E_EDGES + e] : (e - E_EDGES);
  const float a = expf(lg[tid] - mx[d * 8 + hh]);
  lg[tid] = a;
  atomicAdd(&den[d * 8 + hh], a);
}

// ---------------------------------------------------------------- edge pass 3
// gat[dst] += (a/den[dst]) * xl[src]
__global__ void edge_scatter_kernel(const float* __restrict__ lg,
                                    const float* __restrict__ den,
                                    const float* __restrict__ xl,
                                    const int* __restrict__ ei,
                                    float* __restrict__ gat) {
  int tid = blockIdx.x * blockDim.x + threadIdx.x;
  if (tid >= ET_EDGES * 8) return;
  const int e = tid >> 3, hh = tid & 7;
  const int s = (e < E_EDGES) ? ei[e]           : (e - E_EDGES);
  const int d = (e < E_EDGES) ? ei[E_EDGES + e] : (e - E_EDGES);
  const float w = lg[tid] / (den[d * 8 + hh] + 1e-16f);
  const float* pl = xl + (size_t)s * HC + hh * 16;
  float*       po = gat + (size_t)d * HC + hh * 16;
#pragma unroll
  for (int q = 0; q < 16; ++q) atomicAdd(&po[q], w * pl[q]);
}

// ---------------------------------------------------------------- batchnorm
__global__ __launch_bounds__(128) void bn_stats_kernel(
    const float* __restrict__ x, float* __restrict__ sum,
    float* __restrict__ sq) {
  const int c = threadIdx.x;
  const int row0 = blockIdx.x * 128;
  float s = 0.0f, q = 0.0f;
  for (int i = 0; i < 128; ++i) {
    const float v = x[(size_t)(row0 + i) * HC + c];
    s += v; q += v * v;
  }
  atomicAdd(&sum[c], s);
  atomicAdd(&sq[c], q);
}

// y = relu( (x-mu)*rsqrt(var+eps)*g + b  [+ res] )
// (GAT output bias bo cancels exactly inside BN and is omitted.)
__global__ void bn_apply_kernel(const float* __restrict__ x,
                                const float* __restrict__ sum,
                                const float* __restrict__ sq,
                                const float* __restrict__ gamma,
                                const float* __restrict__ beta,
                                const float* __restrict__ res,
                                float* __restrict__ out) {
  int i = blockIdx.x * blockDim.x + threadIdx.x;
  if (i >= N_NODES * HC) return;
  const int c = i & 127;
  const float inv = 1.0f / (float)N_NODES;
  const float mu  = sum[c] * inv;
  const float var = sq[c] * inv - mu * mu;
  const float rs  = rsqrtf(var + BN_EPS);
  float v = (x[i] - mu) * rs * gamma[c] + beta[c];
  if (res) v += res[i];
  out[i] = v > 0.0f ? v : 0.0f;
}

// ---------------------------------------------------------------- pool + MLP
__global__ __launch_bounds__(160) void pool_kernel(
    const float* __restrict__ h, const float* __restrict__ gf,
    float* __restrict__ z) {
  const int g = blockIdx.x, t = threadIdx.x;
  if (t < 128) {
    float s = 0.0f;
    for (int i = 0; i < NPG; ++i) s += h[(size_t)(g * NPG + i) * HC + t];
    z[g * (HC + GF_DIM) + t] = s * (1.0f / (float)NPG);
  } else if (t < 128 + GF_DIM) {
    z[g * (HC + GF_DIM) + t] = gf[g * GF_DIM + (t - 128)];
  }
}

__global__ void mlp_kernel(const float* __restrict__ in,
                           const float* __restrict__ W,
                           const float* __restrict__ b,
                           float* __restrict__ out,
                           int din, int dout, int relu) {
  int i = blockIdx.x * blockDim.x + threadIdx.x;
  if (i >= G_GR * dout) return;
  const int g = i / dout, o = i % dout;
  float s = b[o];
  for (int k = 0; k < din; ++k) s += in[g * din + k] * W[k * dout + o];
  if (relu) s = s > 0.0f ? s : 0.0f;
  out[i] = s;
}

// ---------------------------------------------------------------- launcher
extern "C" void kernel_launch(void* const* d_in, const int* in_sizes, int n_in,
                              void* d_out, int out_size, void* d_ws, size_t ws_size,
                              hipStream_t stream) {
  (void)in_sizes; (void)n_in; (void)out_size; (void)ws_size;

  const float* x  = (const float*)d_in[0];
  const int*   ei = (const int*)d_in[1];
  // d_in[2] = batch: nodes are contiguous per graph, index computed directly
  const float* gf = (const float*)d_in[3];
  const float* We = (const float*)d_in[4];
  const float* be = (const float*)d_in[5];

  float* ws  = (float*)d_ws;
  float* hA  = ws; ws += (size_t)N_NODES * HC;
  float* hB  = ws; ws += (size_t)N_NODES * HC;
  float* xl  = ws; ws += (size_t)N_NODES * HC;
  float* xr  = ws; ws += (size_t)N_NODES * HC;
  float* gat = ws; ws += (size_t)N_NODES * HC;
  float* lg  = ws; ws += (size_t)ET_EDGES * 8;
  float* mx  = ws; ws += (size_t)N_NODES * 8;
  float* den = ws; ws += (size_t)N_NODES * 8;
  float* bsum= ws; ws += 128;
  float* bsq = ws; ws += 128;
  float* z   = ws; ws += G_GR * (HC + GF_DIM);
  float* z1  = ws; ws += G_GR * 128;
  float* z2  = ws; ws += G_GR * 64;

  embed_kernel<<<(N_NODES * 64 + 255) / 256, 256, 0, stream>>>(x, We, be, hA);

  float* hin = hA;
  float* hout = hB;
  const int egrid = (ET_EDGES * 8 + 255) / 256;

  for (int l = 0; l < 4; ++l) {
    const float* Wl  = (const float*)d_in[6 + l * 8 + 0];
    const float* bl  = (const float*)d_in[6 + l * 8 + 1];
    const float* Wr  = (const float*)d_in[6 + l * 8 + 2];
    const float* br  = (const float*)d_in[6 + l * 8 + 3];
    const float* att = (const float*)d_in[6 + l * 8 + 4];
    // d_in[6+l*8+5] = bo: cancels inside BatchNorm (mean shift), omitted
    const float* gg  = (const float*)d_in[6 + l * 8 + 6];
    const float* bb  = (const float*)d_in[6 + l * 8 + 7];

    if (l == 0) {
      wmma_gemm_kernel<64><<<N_NODES / 128, 256, 0, stream>>>(hin, Wl, bl, xl);
      wmma_gemm_kernel<64><<<N_NODES / 128, 256, 0, stream>>>(hin, Wr, br, xr);
    } else {
      wmma_gemm_kernel<128><<<N_NODES / 128, 256, 0, stream>>>(hin, Wl, bl, xl);
      wmma_gemm_kernel<128><<<N_NODES / 128, 256, 0, stream>>>(hin, Wr, br, xr);
    }

    fill_kernel<<<(N_NODES * 8 + 255) / 256, 256, 0, stream>>>(mx, -3.402823466e38f, N_NODES * 8);
    fill_kernel<<<(N_NODES * 8 + 255) / 256, 256, 0, stream>>>(den, 0.0f, N_NODES * 8);
    fill_kernel<<<(N_NODES * HC + 255) / 256, 256, 0, stream>>>(gat, 0.0f, N_NODES * HC);
    fill_kernel<<<1, 256, 0, stream>>>(bsum, 0.0f, 128);
    fill_kernel<<<1, 256, 0, stream>>>(bsq, 0.0f, 128);

    edge_logit_kernel<<<egrid, 256, 0, stream>>>(xl, xr, att, ei, lg, mx);
    edge_exp_kernel<<<egrid, 256, 0, stream>>>(lg, mx, den, ei);
    edge_scatter_kernel<<<egrid, 256, 0, stream>>>(lg, den, xl, ei, gat);

    bn_stats_kernel<<<N_NODES / 128, 128, 0, stream>>>(gat, bsum, bsq);
    bn_apply_kernel<<<(N_NODES * HC + 255) / 256, 256, 0, stream>>>(
        gat, bsum, bsq, gg, bb, (l == 0) ? nullptr : hin, hout);

    float* t = hin; hin = hout; hout = t;
  }

  pool_kernel<<<G_GR, 160, 0, stream>>>(hin, gf, z);
  mlp_kernel<<<(G_GR * 128 + 255) / 256, 256, 0, stream>>>(
      z, (const float*)d_in[38], (const float*)d_in[39], z1, HC + GF_DIM, 128, 1);
  mlp_kernel<<<(G_GR * 64 + 255) / 256, 256, 0, stream>>>(
      z1, (const float*)d_in[40], (const float*)d_in[41], z2, 128, 64, 1);
  mlp_kernel<<<1, 256, 0, stream>>>(
      z2, (const float*)d_in[42], (const float*)d_in[43], (float*)d_out, 64, 1, 0);
}